// QPChargeNormalization_76063870812743
// MI455X (gfx1250) — compile-verified
//
#include <hip/hip_runtime.h>

typedef float v2f __attribute__((ext_vector_type(2)));
typedef float v4f __attribute__((ext_vector_type(4)));
typedef float v8f __attribute__((ext_vector_type(8)));

#define NROW4   1024   // 4096 floats per row per array, as float4
#define NATOMS   256

// Full 32-lane (wave32) sum via one V_WMMA_F32_16X16X4_F32.
// A layout (ISA 7.12.2, 32-bit A 16x4): lane m (0-15): VGPR0=A[m][0], VGPR1=A[m][1];
// lane m+16: VGPR0=A[m][2], VGPR1=A[m][3].  We place x in VGPR0, 0 in VGPR1, B=ones:
//   D[m][n] = A[m][0] + A[m][2] = x_m + x_{m+16}   (zero padding makes this
//   independent of B's internal lane mapping since every B entry is 1).
// D layout: VGPR r holds M=r (lanes 0-15) and M=r+8 (lanes 16-31), so summing all
// 8 D VGPRs per lane and adding the lane^16 partner gives the full 32-lane total.
__device__ __forceinline__ float wave_reduce_wmma(float x) {
  v2f a;    a.x = x;    a.y = 0.0f;
  v2f ones; ones.x = 1.0f; ones.y = 1.0f;
  v8f c = {};
  v8f d = __builtin_amdgcn_wmma_f32_16x16x4_f32(
      /*neg_a=*/false, a, /*neg_b=*/false, ones,
      /*c_mod=*/(short)0, c, /*reuse_a=*/false, /*reuse_b=*/false);
  float s = ((d[0] + d[1]) + (d[2] + d[3])) + ((d[4] + d[5]) + (d[6] + d[7]));
  s += __shfl_xor(s, 16, 32);
  return s;  // every lane holds the wave sum
}

__global__ __launch_bounds__(256)
void qp_charge_norm_kernel(const v4f* __restrict__ ci_g,
                           const v4f* __restrict__ ca_g,
                           const v4f* __restrict__ qi_g,
                           const v4f* __restrict__ qa_g,
                           const float* __restrict__ charge,
                           v4f* __restrict__ oi_g,
                           v4f* __restrict__ oa_g) {
  const int b = blockIdx.x;
  const int t = threadIdx.x;                 // 0..255, 8 waves
  const size_t base = (size_t)b * NROW4;

  // ---- Phase 1: single streaming read, everything register-resident ----
  v4f ci[4], ca[4], qi[4], qa[4];
#pragma unroll
  for (int i = 0; i < 4; ++i) {
    const size_t idx = base + (size_t)(t + i * 256);
    ci[i] = __builtin_nontemporal_load(&ci_g[idx]);
    qi[i] = __builtin_nontemporal_load(&qi_g[idx]);
    ca[i] = __builtin_nontemporal_load(&ca_g[idx]);
    qa[i] = __builtin_nontemporal_load(&qa_g[idx]);
  }
  const float ch = __builtin_nontemporal_load(&charge[(size_t)b * NATOMS + t]);

  float s_qc = 0.0f, s_qq = 0.0f;
#pragma unroll
  for (int i = 0; i < 4; ++i) {
    s_qc += qi[i].x * ci[i].x + qi[i].y * ci[i].y +
            qi[i].z * ci[i].z + qi[i].w * ci[i].w;
    s_qc += qa[i].x * ca[i].x + qa[i].y * ca[i].y +
            qa[i].z * ca[i].z + qa[i].w * ca[i].w;
    s_qq += qi[i].x * qi[i].x + qi[i].y * qi[i].y +
            qi[i].z * qi[i].z + qi[i].w * qi[i].w;
    s_qq += qa[i].x * qa[i].x + qa[i].y * qa[i].y +
            qa[i].z * qa[i].z + qa[i].w * qa[i].w;
  }

  // ---- Wave-level reductions on the matrix pipe (EXEC all ones here) ----
  const float w_qc = wave_reduce_wmma(s_qc);
  const float w_qq = wave_reduce_wmma(s_qq);
  const float w_Q  = wave_reduce_wmma(ch);   // 256 charges == 1 per thread

  // ---- Cross-wave reduction: 8 partials x 3 quantities through LDS ----
  __shared__ float red[8][4];
  const int wv = t >> 5;
  const int ln = t & 31;
  if (ln == 0) { red[wv][0] = w_qc; red[wv][1] = w_qq; red[wv][2] = w_Q; }
  __syncthreads();
  float qc = 0.0f, qq = 0.0f, Qs = 0.0f;
#pragma unroll
  for (int w = 0; w < 8; ++w) {
    qc += red[w][0]; qq += red[w][1]; Qs += red[w][2];
  }
  const float hu = (Qs - qc) / qq;           // u/2 = (Q - q^T c) / (q^T q)

  // ---- Phase 2: update from registers, single streaming write ----
#pragma unroll
  for (int i = 0; i < 4; ++i) {
    const size_t idx = base + (size_t)(t + i * 256);
    v4f oi, oa;
    oi.x = ci[i].x + hu * qi[i].x;  oi.y = ci[i].y + hu * qi[i].y;
    oi.z = ci[i].z + hu * qi[i].z;  oi.w = ci[i].w + hu * qi[i].w;
    oa.x = ca[i].x + hu * qa[i].x;  oa.y = ca[i].y + hu * qa[i].y;
    oa.z = ca[i].z + hu * qa[i].z;  oa.w = ca[i].w + hu * qa[i].w;
    __builtin_nontemporal_store(oi, &oi_g[idx]);
    __builtin_nontemporal_store(oa, &oa_g[idx]);
  }
}

extern "C" void kernel_launch(void* const* d_in, const int* in_sizes, int n_in,
                              void* d_out, int out_size, void* d_ws, size_t ws_size,
                              hipStream_t stream) {
  (void)n_in; (void)out_size; (void)d_ws; (void)ws_size;
  const v4f* ci = (const v4f*)d_in[0];  // c_iso    [B,256,16]
  const v4f* ca = (const v4f*)d_in[1];  // c_aniso  [B,512,8]
  const v4f* qi = (const v4f*)d_in[2];  // int_iso  [B,256,16]
  const v4f* qa = (const v4f*)d_in[3];  // int_aniso[B,512,8]
  const float* ch = (const float*)d_in[4];   // charge [B,256]

  const int B = in_sizes[4] / NATOMS;        // 4096

  float* out = (float*)d_out;                // [sol_iso | sol_aniso] flat
  v4f* oi = (v4f*)out;
  v4f* oa = (v4f*)(out + (size_t)B * 4096);

  qp_charge_norm_kernel<<<B, 256, 0, stream>>>(ci, ca, qi, qa, ch, oi, oa);
}